// PointsMatcher_45423574122961
// MI455X (gfx1250) — compile-verified
//
#include <hip/hip_runtime.h>

typedef float v2f __attribute__((ext_vector_type(2)));
typedef float v8f __attribute__((ext_vector_type(8)));

#ifndef __has_builtin
#define __has_builtin(x) 0
#endif

#if __has_builtin(__builtin_amdgcn_wmma_f32_16x16x4_f32)
#define HAVE_WMMA4 1
#else
#define HAVE_WMMA4 0
#endif

#if __has_builtin(__builtin_amdgcn_global_load_async_to_lds_b32) && \
    __has_builtin(__builtin_amdgcn_s_wait_asynccnt)
#define HAVE_ASYNC 1
#else
#define HAVE_ASYNC 0
#endif

#define NGT     200
#define NGT_PAD 208
#define NPTS    5456   // 64*64+32*32+16*16+8*8+4*4
#define NTILES  341    // NPTS/16
#define INFC    1e16f

typedef __attribute__((address_space(1))) int  gint_t;
typedef __attribute__((address_space(3))) int  lint_t;

// d_out float layout (element offsets):
//  bt0 @0 (131072), bt1 @131072 (32768), bt2 @163840 (8192),
//  bt3 @172032 (2048), bt4 @174080 (512),
//  lab0 @174592, lab1 @207360, lab2 @215552, lab3 @217600, lab4 @218112
__global__ __launch_bounds__(256)
void fcos_match_kernel(const float* __restrict__ gts, float* __restrict__ out)
{
    __shared__ __align__(16) float sg[NGT_PAD * 8]; // x1,y1,x2,y2,cls,area,pad,pad
    const int tid = threadIdx.x;
    const int b   = blockIdx.y;
    const float* gb = gts + (size_t)b * (NGT * 5);

    // ---- stage GT rows of this batch into LDS (async TDM-style path) ----
#if HAVE_ASYNC
    for (int i = tid; i < NGT * 5; i += 256) {
        int g = i / 5, c = i - g * 5;
        __builtin_amdgcn_global_load_async_to_lds_b32(
            (gint_t*)(gb + i),
            (lint_t*)(&sg[g * 8 + c]),
            0, 0);
    }
    __builtin_amdgcn_s_wait_asynccnt(0);
#else
    for (int i = tid; i < NGT * 5; i += 256) {
        int g = i / 5, c = i - g * 5;
        sg[g * 8 + c] = gb[i];
    }
#endif
    __syncthreads();
    if (tid < NGT_PAD) {
        if (tid < NGT) {
            float x1 = sg[tid*8+0], y1 = sg[tid*8+1];
            float x2 = sg[tid*8+2], y2 = sg[tid*8+3];
            sg[tid*8+5] = (x2 - x1) * (y2 - y1);
        } else { // pad: cls=-1 -> never valid
            sg[tid*8+0] = 0.f; sg[tid*8+1] = 0.f; sg[tid*8+2] = 0.f;
            sg[tid*8+3] = 0.f; sg[tid*8+4] = -1.f; sg[tid*8+5] = 0.f;
        }
    }
    __syncthreads();

    // one wave == one 16-point tile (wave-uniform guard keeps EXEC full for WMMA)
    const int t = blockIdx.x * 8 + (tid >> 5);
    if (t >= NTILES) return;

    int   lbase, log2W, PS, btb, labb;
    float invs, rr0s, rr1s;
    const int base = t << 4;
    if      (base < 4096) { lbase=0;    log2W=6; PS=4096; invs=0.125f;     rr0s=-0.125f; rr1s=8.f;        btb=0;      labb=174592; }
    else if (base < 5120) { lbase=4096; log2W=5; PS=1024; invs=0.0625f;    rr0s=4.f;     rr1s=8.f;        btb=131072; labb=207360; }
    else if (base < 5376) { lbase=5120; log2W=4; PS=256;  invs=0.03125f;   rr0s=4.f;     rr1s=8.f;        btb=163840; labb=215552; }
    else if (base < 5440) { lbase=5376; log2W=3; PS=64;   invs=0.015625f;  rr0s=4.f;     rr1s=8.f;        btb=172032; labb=217600; }
    else                  { lbase=5440; log2W=2; PS=16;   invs=0.0078125f; rr0s=4.f;     rr1s=7.8125e13f; btb=174080; labb=218112; }
    const int Wm1       = (1 << log2W) - 1;
    const int localbase = base - lbase;

    const int lane = tid & 31;
    const int half = lane >> 4;   // lane half selects K-block / M-block
    const int lr   = lane & 15;   // N (gt within tile) for B/C/D; M for A low half

    float best[8]; int bidx[8];
    #pragma unroll
    for (int e = 0; e < 8; ++e) { best[e] = INFC; bidx[e] = 0; }

#if HAVE_WMMA4
    // A (16x4): row M = [px, py, 1, 0]. Lane layout: VGPR r, half h -> K = r + 2h.
    const int   pA = localbase + lr;
    const float px = (float)(pA & Wm1);
    const float py = (float)(pA >> log2W);
    v2f A;
    A.x = half ? 1.0f : px;
    A.y = half ? 0.0f : py;
#endif

    for (int g = 0; g < 13; ++g) {
        const int idx = (g << 4) + lr;                 // gt handled by this lane (N = lr)
        const float4 bb = *reinterpret_cast<const float4*>(&sg[idx * 8]);
        const float clsv = sg[idx * 8 + 4];
        const float area = sg[idx * 8 + 5];
        const float sx1 = bb.x * invs, sy1 = bb.y * invs;
        const float sx2 = bb.z * invs, sy2 = bb.w * invs;

        float dl[8], dt[8], dr[8], db[8];
#if HAVE_WMMA4
        // B (4x16) columns: l:[1,0,-sx1,0] t:[0,1,-sy1,0] r:[-1,0,sx2,0] b:[0,-1,sy2,0]
        v2f Bl, Bt, Br, Bb;
        Bl.x = half ? -sx1 : 1.0f;   Bl.y = 0.0f;
        Bt.x = half ? -sy1 : 0.0f;   Bt.y = half ? 0.0f : 1.0f;
        Br.x = half ?  sx2 : -1.0f;  Br.y = 0.0f;
        Bb.x = half ?  sy2 : 0.0f;   Bb.y = half ? 0.0f : -1.0f;
        v8f c0 = {0.f, 0.f, 0.f, 0.f, 0.f, 0.f, 0.f, 0.f};
        v8f Dl = __builtin_amdgcn_wmma_f32_16x16x4_f32(false, A, false, Bl, (short)0, c0, false, false);
        v8f Dt = __builtin_amdgcn_wmma_f32_16x16x4_f32(false, A, false, Bt, (short)0, c0, false, false);
        v8f Dr = __builtin_amdgcn_wmma_f32_16x16x4_f32(false, A, false, Br, (short)0, c0, false, false);
        v8f Db = __builtin_amdgcn_wmma_f32_16x16x4_f32(false, A, false, Bb, (short)0, c0, false, false);
        #pragma unroll
        for (int e = 0; e < 8; ++e) { dl[e]=Dl[e]; dt[e]=Dt[e]; dr[e]=Dr[e]; db[e]=Db[e]; }
#else
        #pragma unroll
        for (int e = 0; e < 8; ++e) {
            const int   pe  = localbase + e + (half << 3);
            const float pxe = (float)(pe & Wm1);
            const float pye = (float)(pe >> log2W);
            dl[e] = pxe - sx1; dt[e] = pye - sy1;
            dr[e] = sx2 - pxe; db[e] = sy2 - pye;
        }
#endif
        #pragma unroll
        for (int e = 0; e < 8; ++e) {
            const float mn = fminf(fminf(dl[e], dt[e]), fminf(dr[e], db[e]));
            const float mx = fmaxf(fmaxf(dl[e], dt[e]), fmaxf(dr[e], db[e]));
            const bool ok = (mn > 0.f) && (mx >= rr0s) && (mx <= rr1s) && (clsv >= 0.f);
            const float a = ok ? area : INFC;
            const bool upd = a < best[e];     // strict '<' keeps first min (argmin rule)
            best[e] = upd ? a   : best[e];
            bidx[e] = upd ? idx : bidx[e];
        }
    }

    // argmin across 16 gts (lane direction within each 16-lane half)
    #pragma unroll
    for (int m = 1; m < 16; m <<= 1) {
        #pragma unroll
        for (int e = 0; e < 8; ++e) {
            const float ob = __shfl_xor(best[e], m, 32);
            const int   oi = __shfl_xor(bidx[e], m, 32);
            const bool take = (ob < best[e]) || ((ob == best[e]) && (oi < bidx[e]));
            best[e] = take ? ob : best[e];
            bidx[e] = take ? oi : bidx[e];
        }
    }

    // lanes with lr<8 (in both halves) each write one of the 16 points: M = lr + 8*half
    if (lr < 8) {
        float mb = best[0]; int mi = bidx[0];
        #pragma unroll
        for (int e = 1; e < 8; ++e) {
            const bool c = (lr == e);
            mb = c ? best[e] : mb;
            mi = c ? bidx[e] : mi;
        }
        const int   pl = localbase + lr + (half << 3);
        const float x  = (float)(pl & Wm1);
        const float y  = (float)(pl >> log2W);
        const float4 bbw = *reinterpret_cast<const float4*>(&sg[mi * 8]);
        const float clsw = sg[mi * 8 + 4];
        float4 o;
        o.x = x - bbw.x * invs;
        o.y = y - bbw.y * invs;
        o.z = bbw.z * invs - x;
        o.w = bbw.w * invs - y;
        const float label = (mb == INFC) ? 80.0f : clsw;
        const size_t pos = (size_t)b * PS + pl;
        *reinterpret_cast<float4*>(out + btb + pos * 4) = o;
        out[labb + pos] = label;
    }
}

extern "C" void kernel_launch(void* const* d_in, const int* in_sizes, int n_in,
                              void* d_out, int out_size, void* d_ws, size_t ws_size,
                              hipStream_t stream) {
    const float* gts = (const float*)d_in[5];   // feats d_in[0..4] unused (shape-only)
    float* out = (float*)d_out;
    const int B = in_sizes[5] / (NGT * 5);      // = 8
    dim3 grid((NTILES + 7) / 8, B);
    fcos_match_kernel<<<grid, dim3(256), 0, stream>>>(gts, out);
}